// EdgeScoreDotProductGAT_4698694221870
// MI455X (gfx1250) — compile-verified
//
#include <hip/hip_runtime.h>

#define N_NODES   100000
#define E_EDGES   3200000
#define D_FEAT    256
#define H_HEADS   4
#define DH_HEAD   32
#define NEG_SLOPE 0.2f

typedef __attribute__((ext_vector_type(2))) float v2f;
typedef __attribute__((ext_vector_type(8))) float v8f;

// ---------------------------------------------------------------------------
// Kernel 1: fold attention vector a into Wq / Wk, producing zero-padded
// 16 x 256 weight tiles (rows 4..15 = 0) so the WMMA A-operand can be loaded
// unconditionally by all 32 lanes.
//   vq[h][d] = sum_k a[h][k]    * Wq[h*32+k][d]
//   vk[h][d] = sum_k a[h][32+k] * Wk[h*32+k][d]
// grid (16, 2), block 256: blockIdx.x = padded row, blockIdx.y = side.
// ---------------------------------------------------------------------------
__global__ void gat_fold_weights(const float* __restrict__ Wq,
                                 const float* __restrict__ Wk,
                                 const float* __restrict__ a,
                                 float* __restrict__ vq,
                                 float* __restrict__ vk) {
    const int hp  = blockIdx.x;          // 0..15 (padded head row)
    const int d   = threadIdx.x;         // 0..255
    const int dst = blockIdx.y;          // 0 = q side, 1 = k side

    const float* __restrict__ W  = dst ? Wk : Wq;
    const float* __restrict__ av = a + (dst ? DH_HEAD : 0);
    float* __restrict__ o        = dst ? vk : vq;

    float s = 0.0f;
    if (hp < H_HEADS) {
        const float* __restrict__ wrow = W + (size_t)(hp * DH_HEAD) * D_FEAT + d;
        const float* __restrict__ arow = av + hp * (2 * DH_HEAD);
#pragma unroll
        for (int k = 0; k < DH_HEAD; ++k)
            s = fmaf(arow[k], wrow[(size_t)k * D_FEAT], s);
    }
    o[hp * D_FEAT + d] = s;              // rows 4..15 get exact zeros
}

// ---------------------------------------------------------------------------
// Kernel 2: node scores via V_WMMA_F32_16X16X4_F32 (full fp32 matrix pipe).
//   D(16x16) = A(16x4) x B(4x16) + C,  M = head (padded), N = node, K = dims
// One wave per 16-node tile, 64 accumulating WMMAs sweep K = 256.
// A-layout (32b A 16x4): v0 = K{0|2}, v1 = K{1|3}; M = lane&15.
// B-layout (32b B 4x16): v0 = row K{0|2}, v1 = row K{1|3}; N = lane&15.
// D-layout: VGPR r, lanes 0-15 -> M=r, N=lane  =>  lane L holds
// el[node L][head 0..3] in acc[0..3]  ->  one coalesced float4 store.
// grid (6250, 2), block 32.
// ---------------------------------------------------------------------------
__global__ void gat_node_scores(const float* __restrict__ h_src,
                                const float* __restrict__ h_dst,
                                const float* __restrict__ vq,
                                const float* __restrict__ vk,
                                float* __restrict__ el,
                                float* __restrict__ er) {
    const int lane = threadIdx.x & 31;
    const int tile = blockIdx.x;                 // 16-node tile
    const int side = blockIdx.y;                 // 0 = src/q, 1 = dst/k

    const float* __restrict__ h = side ? h_dst : h_src;
    const float* __restrict__ w = side ? vk : vq;
    float* __restrict__ o       = side ? er : el;

    const int half = lane >> 4;                  // 0: K+{0,1}, 1: K+{2,3}
    const int lid  = lane & 15;                  // M (head row) / N (node)

    const float* __restrict__ aptr = w + (size_t)lid * D_FEAT + 2 * half;
    const float* __restrict__ bptr =
        h + ((size_t)tile * 16 + lid) * D_FEAT + 2 * half;

    v8f acc = {0.f, 0.f, 0.f, 0.f, 0.f, 0.f, 0.f, 0.f};

#pragma unroll 8
    for (int kt = 0; kt < D_FEAT / 4; ++kt) {
        v2f amat = *(const v2f*)(aptr + kt * 4);   // folded weights (hot, L2)
        v2f bmat = *(const v2f*)(bptr + kt * 4);   // node features (stream)
        // 8 args: (neg_a, A, neg_b, B, c_mod, C, reuse_a, reuse_b)
        acc = __builtin_amdgcn_wmma_f32_16x16x4_f32(
            false, amat, false, bmat, (short)0, acc, false, false);
    }

    if (half == 0) {                             // heads 0..3 live in lanes 0-15
        float4 r = make_float4(acc[0], acc[1], acc[2], acc[3]);
        *(float4*)(o + ((size_t)tile * 16 + lid) * H_HEADS) = r;
    }
}

// ---------------------------------------------------------------------------
// Kernel 3: edge gather + leaky ReLU. el/er are 1.6 MB each -> L2-resident
// gathers; output store is fully coalesced float4.
// grid 12500, block 256 (E = 3.2M exact).
// ---------------------------------------------------------------------------
__device__ __forceinline__ float lrelu(float x) {
    return x >= 0.0f ? x : NEG_SLOPE * x;
}

__global__ void gat_edge_scores(const int* __restrict__ src_idx,
                                const int* __restrict__ dst_idx,
                                const float* __restrict__ el,
                                const float* __restrict__ er,
                                float* __restrict__ out) {
    const long e = (long)blockIdx.x * blockDim.x + threadIdx.x;
    if (e >= E_EDGES) return;
    const int s = src_idx[e];
    const int d = dst_idx[e];
    float4 a = *(const float4*)(el + (size_t)s * H_HEADS);
    float4 b = *(const float4*)(er + (size_t)d * H_HEADS);
    float4 r;
    r.x = lrelu(a.x + b.x);
    r.y = lrelu(a.y + b.y);
    r.z = lrelu(a.z + b.z);
    r.w = lrelu(a.w + b.w);
    *(float4*)(out + (size_t)e * H_HEADS) = r;
}

// ---------------------------------------------------------------------------
// Launch. Inputs: 0 h_src, 1 h_dst, 2 src_idx, 3 dst_idx, 4 Wq, 5 Wk, 6 a.
// ws layout: [0,16KB) vq | [16KB,32KB) vk | el (1.6MB) | er (1.6MB).
// ---------------------------------------------------------------------------
extern "C" void kernel_launch(void* const* d_in, const int* in_sizes, int n_in,
                              void* d_out, int out_size, void* d_ws, size_t ws_size,
                              hipStream_t stream) {
    const float* h_src = (const float*)d_in[0];
    const float* h_dst = (const float*)d_in[1];
    const int*   s_idx = (const int*)d_in[2];
    const int*   d_idx = (const int*)d_in[3];
    const float* Wq    = (const float*)d_in[4];
    const float* Wk    = (const float*)d_in[5];
    const float* a     = (const float*)d_in[6];
    float*       out   = (float*)d_out;

    float* ws = (float*)d_ws;
    float* vq = ws;                        // 16*256 floats
    float* vk = ws + 16 * D_FEAT;          // 16*256 floats
    float* el = ws + 32 * D_FEAT;          // N*4 floats
    float* er = el + (size_t)N_NODES * H_HEADS;

    gat_fold_weights<<<dim3(16, 2), 256, 0, stream>>>(Wq, Wk, a, vq, vk);

    gat_node_scores<<<dim3(N_NODES / 16, 2), 32, 0, stream>>>(
        h_src, h_dst, vq, vk, el, er);

    gat_edge_scores<<<dim3(E_EDGES / 256), 256, 0, stream>>>(
        s_idx, d_idx, el, er, out);
}